// KGCR_11544872091694
// MI455X (gfx1250) — compile-verified
//
#include <hip/hip_runtime.h>
#include <hip/hip_bf16.h>

#define B_ 256
#define L_ 50
#define N_ 100000
#define D_ 128

typedef __attribute__((ext_vector_type(2))) float v2f;
typedef __attribute__((ext_vector_type(8))) float v8f;

// Workspace layout in floats
#define WS_Q1  0
#define WS_Q2  128
#define WS_C   256
#define WS_U   272                 // B_*D_ = 32768 floats
#define WS_LSE (272 + B_ * D_)     // B_ floats

// ---------------------------------------------------------------------------
// Kernel A: q1 = W1^T q, q2 = W2^T q, c = b_att * sum(q).  1 block, 128 thr.
// ---------------------------------------------------------------------------
__global__ void kA_precompute(const float* __restrict__ W1,
                              const float* __restrict__ W2,
                              const float* __restrict__ q,
                              const float* __restrict__ b_att,
                              float* __restrict__ w) {
  int d = threadIdx.x;                  // 0..127
  float s1 = 0.f, s2 = 0.f, qs = 0.f;
  for (int e = 0; e < D_; ++e) {
    float qe = q[e];
    s1 += W1[e * D_ + d] * qe;
    s2 += W2[e * D_ + d] * qe;
    qs += qe;
  }
  w[WS_Q1 + d] = s1;
  w[WS_Q2 + d] = s2;
  if (d == 0) w[WS_C] = b_att[0] * qs;
}

// ---------------------------------------------------------------------------
// Kernel B: per-batch  v = E[seeds]+pe,  att[l] = v·q1 + vn·q2 + c,
//           u[b,:] = sum_l att[l] * v[l,:].   B blocks, 128 threads.
// ---------------------------------------------------------------------------
__global__ void kB_batch_u(const int* __restrict__ seeds,
                           const float* __restrict__ E,
                           const float* __restrict__ pe,
                           const float* __restrict__ w,
                           float* __restrict__ u) {
  __shared__ float vbuf[L_][D_];   // 25.6 KB
  __shared__ float red[D_];
  __shared__ float atts[L_];
  __shared__ float q1s[D_];
  __shared__ float s2sh;

  int b = blockIdx.x;
  int d = threadIdx.x;
  float q1d = w[WS_Q1 + d];
  float q2d = w[WS_Q2 + d];
  q1s[d] = q1d;

  for (int l = 0; l < L_; ++l) {
    int s = seeds[b * L_ + l];
    vbuf[l][d] = E[(size_t)s * D_ + d] + pe[l * D_ + d];
  }
  __syncthreads();

  // s2 = dot(vn, q2) + c  (block reduction over 128 lanes)
  red[d] = vbuf[L_ - 1][d] * q2d;
  __syncthreads();
  for (int off = D_ / 2; off > 0; off >>= 1) {
    if (d < off) red[d] += red[d + off];
    __syncthreads();
  }
  if (d == 0) s2sh = red[0] + w[WS_C];
  __syncthreads();

  // att[l] for l < 50, one per thread
  if (d < L_) {
    float a = 0.f;
    for (int k = 0; k < D_; ++k) a += vbuf[d][k] * q1s[k];
    atts[d] = a + s2sh;
  }
  __syncthreads();

  float acc = 0.f;
  for (int l = 0; l < L_; ++l) acc += atts[l] * vbuf[l][d];
  u[b * D_ + d] = acc;
}

// ---------------------------------------------------------------------------
// Kernel C: scores[b,n] = sum_d u[b,d]*E[n,d] + out_bias[n]
// One wave (32 threads) computes one 16x16 tile via 32 chained
// V_WMMA_F32_16X16X4_F32.  Grid: (N/16 = 6250, B/16 = 16).
// A layout (16x4 f32): lanes 0-15 -> K=0,1 ; lanes 16-31 -> K=2,3.
// C/D layout: VGPR r of lane -> row = r + 8*(lane>=16), col = lane&15.
// ---------------------------------------------------------------------------
__global__ void kC_gemm(const float* __restrict__ u,
                        const float* __restrict__ E,
                        const float* __restrict__ out_bias,
                        float* __restrict__ scores) {
  int lane = threadIdx.x;
  int hl   = lane >> 4;      // half-wave select
  int idx  = lane & 15;
  int colBase = blockIdx.x * 16;   // n
  int rowBase = blockIdx.y * 16;   // b

  const float* urow = u + (size_t)(rowBase + idx) * D_ + 2 * hl;
  const float* erow = E + (size_t)(colBase + idx) * D_ + 2 * hl;

  v8f acc = {};
#pragma unroll
  for (int kk = 0; kk < D_; kk += 4) {
    v2f a  = *(const v2f*)(urow + kk);
    v2f bb = *(const v2f*)(erow + kk);
    acc = __builtin_amdgcn_wmma_f32_16x16x4_f32(
        /*neg_a=*/false, a, /*neg_b=*/false, bb,
        /*c_mod=*/(short)0, acc, /*reuse_a=*/false, /*reuse_b=*/false);
  }

  float bias = out_bias[colBase + idx];
#pragma unroll
  for (int r = 0; r < 8; ++r) {
    int row = rowBase + r + 8 * hl;
    scores[(size_t)row * N_ + colBase + idx] = acc[r] + bias;
  }
}

// ---------------------------------------------------------------------------
// Kernel D: per-row online log-sum-exp over N=100000.  B blocks, 256 threads.
// ---------------------------------------------------------------------------
__global__ void kD_lse(const float* __restrict__ scores,
                       float* __restrict__ lse) {
  __shared__ float ms[256], ss[256];
  int b = blockIdx.x, t = threadIdx.x;
  float m = -__builtin_inff(), s = 0.f;
  const float* row = scores + (size_t)b * N_;
  for (int n = t; n < N_; n += 256) {
    float x = row[n];
    if (x > m) { s = s * __expf(m - x) + 1.0f; m = x; }
    else       { s += __expf(x - m); }
  }
  ms[t] = m; ss[t] = s;
  __syncthreads();
  for (int off = 128; off > 0; off >>= 1) {
    if (t < off) {
      float m2 = ms[t + off], s2 = ss[t + off];
      float M = fmaxf(ms[t], m2);
      ss[t] = ss[t] * __expf(ms[t] - M) + s2 * __expf(m2 - M);
      ms[t] = M;
    }
    __syncthreads();
  }
  if (t == 0) lse[b] = ms[0] + __logf(ss[0]);
}

// ---------------------------------------------------------------------------
// Kernel E: base_loss = -mean_b( scores[b,labels[b]] - lse[b] ). 1 block.
// ---------------------------------------------------------------------------
__global__ void kE_loss(const float* __restrict__ scores,
                        const int* __restrict__ labels,
                        const float* __restrict__ lse,
                        float* __restrict__ loss_out) {
  __shared__ float red[B_];
  int t = threadIdx.x;
  int lab = labels[t];
  red[t] = scores[(size_t)t * N_ + lab] - lse[t];
  __syncthreads();
  for (int off = B_ / 2; off > 0; off >>= 1) {
    if (t < off) red[t] += red[t + off];
    __syncthreads();
  }
  if (t == 0) loss_out[0] = -red[0] / (float)B_;
}

// ---------------------------------------------------------------------------
extern "C" void kernel_launch(void* const* d_in, const int* in_sizes, int n_in,
                              void* d_out, int out_size, void* d_ws, size_t ws_size,
                              hipStream_t stream) {
  const int*   seeds    = (const int*)d_in[0];
  const int*   labels   = (const int*)d_in[1];
  const float* E        = (const float*)d_in[2];
  const float* W1       = (const float*)d_in[3];
  const float* W2       = (const float*)d_in[4];
  const float* q        = (const float*)d_in[5];
  const float* b_att    = (const float*)d_in[6];
  const float* out_bias = (const float*)d_in[7];
  const float* pe       = (const float*)d_in[8];

  float* scores = (float*)d_out;               // B_*N_ floats
  float* loss   = scores + (size_t)B_ * N_;    // 1 float
  float* w      = (float*)d_ws;
  float* uptr   = w + WS_U;
  float* lse    = w + WS_LSE;

  kA_precompute<<<1, D_, 0, stream>>>(W1, W2, q, b_att, w);
  kB_batch_u<<<B_, D_, 0, stream>>>(seeds, E, pe, w, uptr);
  kC_gemm<<<dim3(N_ / 16, B_ / 16, 1), 32, 0, stream>>>(uptr, E, out_bias, scores);
  kD_lse<<<B_, 256, 0, stream>>>(scores, lse);
  kE_loss<<<1, B_, 0, stream>>>(scores, labels, lse, loss);
}